// HybridQCNN_52467320488413
// MI455X (gfx1250) — compile-verified
//
#include <hip/hip_runtime.h>
#include <math.h>

// ---------------------------------------------------------------------------
// Hybrid QCNN, factorized product-state simulation (all-real arithmetic).
//
// Kernel 1 (qcnn_sim): one wave32 per (batch, half). Each half = qubits
//   {0..7} or {8..15}. Two 4-qubit group states (16 floats each, computed
//   redundantly per lane) are tensor-combined into a 256-dim real state held
//   as 8 floats/lane (lane bits = wire bits b0..b4, element bits = b5..b7).
//   conv2 RY/CNOT gates become shfl_xor butterflies / conditional exchanges.
//   <Z_w> reductions via shfl_xor tree. Writes q[128,8] to workspace.
//
// Kernel 2 (qcnn_head): q[128,8] @ W^T[8,10] + bias via
//   V_WMMA_F32_16X16X4_F32 (two chained k=4 steps, N padded to 16).
// ---------------------------------------------------------------------------

#define PI_F 3.14159265358979323846f

typedef float v2f __attribute__((ext_vector_type(2)));
typedef float v8f __attribute__((ext_vector_type(8)));

// Simulate one 4-qubit group: RY(pi*x_k) encoding on each qubit, then conv1
// filter = RY(w1[k]) on each qubit + CNOT ring (0,1)(1,2)(2,3)(3,0).
// State s[16], index bits (q0 q1 q2 q3), q0 = bit value 8 (MSB).
__device__ __forceinline__ void sim_group(const float* __restrict__ xq,
                                          const float* __restrict__ w1,
                                          float* s) {
    float cs[4], sn[4];
#pragma unroll
    for (int k = 0; k < 4; ++k) {
        sincosf(0.5f * PI_F * xq[k], &sn[k], &cs[k]);
    }
    // product state after encoding RYs (initial |0000>)
#pragma unroll
    for (int i = 0; i < 16; ++i) {
        float v = 1.0f;
#pragma unroll
        for (int k = 0; k < 4; ++k)
            v *= ((i >> (3 - k)) & 1) ? sn[k] : cs[k];
        s[i] = v;
    }
    // conv1: RY(w1[k]) on local wire k
#pragma unroll
    for (int k = 0; k < 4; ++k) {
        float gs, gc;
        sincosf(0.5f * w1[k], &gs, &gc);
        const int str = 8 >> k;
#pragma unroll
        for (int i = 0; i < 16; ++i) {
            if ((i & str) == 0) {
                float a = s[i], b = s[i | str];
                s[i]       = gc * a - gs * b;
                s[i | str] = gs * a + gc * b;
            }
        }
    }
    // conv1 CNOT ring: (c,t) bit-values
    const int cb[4] = {8, 4, 2, 1};
    const int tb[4] = {4, 2, 1, 8};
#pragma unroll
    for (int k = 0; k < 4; ++k) {
#pragma unroll
        for (int i = 0; i < 16; ++i) {
            if ((i & cb[k]) && !(i & tb[k])) {
                float t = s[i];
                s[i] = s[i | tb[k]];
                s[i | tb[k]] = t;
            }
        }
    }
}

__global__ void qcnn_sim(const float* __restrict__ x,     // [128,16]
                         const float* __restrict__ c1w,   // [4]
                         const float* __restrict__ c2w,   // [4]
                         float* __restrict__ q) {         // [128,8]
    const int wave = (blockIdx.x * blockDim.x + threadIdx.x) >> 5;  // 0..255
    const int lane = threadIdx.x & 31;
    const int b    = wave >> 1;
    const int half = wave & 1;

    // two 4-qubit groups of this half (computed redundantly in every lane)
    float v0[16], v1[16];
    sim_group(x + b * 16 + (2 * half) * 4,     c1w, v0);
    sim_group(x + b * 16 + (2 * half + 1) * 4, c1w, v1);

    // tensor product -> 256-dim half state; lane = (b0 b1 b2 b3 b4), e = (b5 b6 b7)
    float psi[8];
    const int i0 = lane >> 1;                    // bits b0..b3
#pragma unroll
    for (int e = 0; e < 8; ++e) {
        const int i1 = ((lane & 1) << 3) | e;    // bits b4..b7
        psi[e] = v0[i0] * v1[i1];
    }

    // conv2 filter on local wires (0,2,4,6) -> bits b0(b16-lane), b2(4), b4(1), b6(elem bit 2)
    float g0s, g0c, g1s, g1c, g2s, g2c, g3s, g3c;
    sincosf(0.5f * c2w[0], &g0s, &g0c);
    sincosf(0.5f * c2w[1], &g1s, &g1c);
    sincosf(0.5f * c2w[2], &g2s, &g2c);
    sincosf(0.5f * c2w[3], &g3s, &g3c);

    // RY wire0 : cross-lane mask 16
#pragma unroll
    for (int e = 0; e < 8; ++e) {
        float p = __shfl_xor(psi[e], 16, 32);
        psi[e] = (lane & 16) ? (g0s * p + g0c * psi[e]) : (g0c * psi[e] - g0s * p);
    }
    // RY wire2 : cross-lane mask 4
#pragma unroll
    for (int e = 0; e < 8; ++e) {
        float p = __shfl_xor(psi[e], 4, 32);
        psi[e] = (lane & 4) ? (g1s * p + g1c * psi[e]) : (g1c * psi[e] - g1s * p);
    }
    // RY wire4 : cross-lane mask 1
#pragma unroll
    for (int e = 0; e < 8; ++e) {
        float p = __shfl_xor(psi[e], 1, 32);
        psi[e] = (lane & 1) ? (g2s * p + g2c * psi[e]) : (g2c * psi[e] - g2s * p);
    }
    // RY wire6 : in-lane, element-bit 2
#pragma unroll
    for (int e = 0; e < 8; ++e) {
        if (!(e & 2)) {
            float a = psi[e], bb = psi[e | 2];
            psi[e]     = g3c * a - g3s * bb;
            psi[e | 2] = g3s * a + g3c * bb;
        }
    }
    // CNOT(0,2): control lane-bit 16, flip lane-bit 4
#pragma unroll
    for (int e = 0; e < 8; ++e) {
        float p = __shfl_xor(psi[e], 4, 32);
        if (lane & 16) psi[e] = p;
    }
    // CNOT(2,4): control lane-bit 4, flip lane-bit 1
#pragma unroll
    for (int e = 0; e < 8; ++e) {
        float p = __shfl_xor(psi[e], 1, 32);
        if (lane & 4) psi[e] = p;
    }
    // CNOT(4,6): control lane-bit 1, flip element-bit 2 (pure local permute)
    if (lane & 1) {
#pragma unroll
        for (int e = 0; e < 8; ++e) {
            if (!(e & 2)) {
                float t = psi[e];
                psi[e] = psi[e | 2];
                psi[e | 2] = t;
            }
        }
    }
    // CNOT(6,0): control element-bit 2, flip lane-bit 16 (symmetric exchange)
#pragma unroll
    for (int e = 0; e < 8; ++e) {
        float p = __shfl_xor(psi[e], 16, 32);
        if (e & 2) psi[e] = p;
    }

    // <Z> on local wires 0,2,4,6 (half state has norm 1, other half marginalizes out)
    float sum = 0.f, s6 = 0.f;
#pragma unroll
    for (int e = 0; e < 8; ++e) {
        float p2 = psi[e] * psi[e];
        sum += p2;
        s6  += (e & 2) ? -p2 : p2;
    }
    float s0 = (lane & 16) ? -sum : sum;
    float s2 = (lane & 4)  ? -sum : sum;
    float s4 = (lane & 1)  ? -sum : sum;
#pragma unroll
    for (int off = 16; off > 0; off >>= 1) {
        s0 += __shfl_xor(s0, off, 32);
        s2 += __shfl_xor(s2, off, 32);
        s4 += __shfl_xor(s4, off, 32);
        s6 += __shfl_xor(s6, off, 32);
    }
    if (lane == 0) {
        float* qq = q + b * 8 + half * 4;
        qq[0] = s0; qq[1] = s2; qq[2] = s4; qq[3] = s6;
    }
}

// Linear head: out[128,10] = q[128,8] @ W^T + bias, via f32 WMMA 16x16x4.
// One wave per 16-row tile (8 waves, single 256-thread block).
__global__ void qcnn_head(const float* __restrict__ q,     // [128,8]
                          const float* __restrict__ W,     // [10,8]
                          const float* __restrict__ bias,  // [10]
                          float* __restrict__ out) {       // [128,10]
    const int wave = threadIdx.x >> 5;   // row tile 0..7
    const int lane = threadIdx.x & 31;
    const int nl   = lane & 15;          // N column (or M row for A)
    const int hi   = lane >> 4;          // 0: K pair {0,1}; 1: K pair {2,3}

    // C seeded with per-column bias (same for every M row)
    const float bn = (nl < 10) ? bias[nl] : 0.0f;
    v8f c;
#pragma unroll
    for (int j = 0; j < 8; ++j) c[j] = bn;

    const int m = wave * 16 + nl;        // A-matrix row for this lane
#pragma unroll
    for (int kk = 0; kk < 2; ++kk) {
        const int kb = kk * 4 + hi * 2;  // K base for this lane's VGPR pair
        // A 16x4: lanes 0-15 -> K={kb,kb+1}; lanes 16-31 -> K shifted by 2 (hi)
        v2f a;
        a.x = q[m * 8 + kb];
        a.y = q[m * 8 + kb + 1];
        // B 4x16 (K x N): B[k][n] = W[n][k]; same K striping as A
        v2f bm;
        bm.x = (nl < 10) ? W[nl * 8 + kb]     : 0.0f;
        bm.y = (nl < 10) ? W[nl * 8 + kb + 1] : 0.0f;
        c = __builtin_amdgcn_wmma_f32_16x16x4_f32(
                /*neg_a=*/false, a, /*neg_b=*/false, bm,
                /*c_mod=*/(short)0, c, /*reuse_a=*/false, /*reuse_b=*/false);
    }

    // D layout: VGPR j -> M = j + 8*hi (within tile), N = nl
    if (nl < 10) {
#pragma unroll
        for (int j = 0; j < 8; ++j) {
            const int mm = wave * 16 + j + 8 * hi;
            out[mm * 10 + nl] = c[j];
        }
    }
}

extern "C" void kernel_launch(void* const* d_in, const int* in_sizes, int n_in,
                              void* d_out, int out_size, void* d_ws, size_t ws_size,
                              hipStream_t stream) {
    const float* x    = (const float*)d_in[0];  // [128,4,4]
    const float* c1w  = (const float*)d_in[1];  // [4]
    const float* c2w  = (const float*)d_in[2];  // [4]
    const float* W    = (const float*)d_in[3];  // [10,8]
    const float* bias = (const float*)d_in[4];  // [10]
    float* out = (float*)d_out;                 // [128,10]
    float* q   = (float*)d_ws;                  // scratch: 128*8 floats

    // 256 waves = 128 batches x 2 independent halves; 8 waves/block
    qcnn_sim<<<32, 256, 0, stream>>>(x, c1w, c2w, q);
    // tiny WMMA GEMM head
    qcnn_head<<<1, 256, 0, stream>>>(q, W, bias, out);
}